// MoEAllReduce_10411000726126
// MI455X (gfx1250) — compile-verified
//
#include <hip/hip_runtime.h>
#include <stdint.h>

// ---------------------------------------------------------------------------
// MoE expert-weighted reduce + residual add + RMSNorm, fused, HBM-bound.
// E=8, T=4096, H=4096, fp32.  One block per token row, 256 threads (8 waves).
// active[e,t,:] is staged through LDS with the CDNA5 async-DMA path
// (global_load_async_to_lds_b128 / s_wait_asynccnt), double-buffered.
// ---------------------------------------------------------------------------

typedef float v4f __attribute__((ext_vector_type(4)));

constexpr int kH      = 4096;
constexpr int kE      = 8;
constexpr int kBlock  = 256;
constexpr int kRow4   = kH / 4;            // 1024 float4 per row
constexpr int kChunks = kRow4 / kBlock;    // 4 float4 chunks per thread

// CDNA5 async copy: each lane copies 16 bytes from global memory at sbase plus g_byte to LDS at lds_byte.
// Tracked by ASYNCcnt (not LOADcnt).
__device__ __forceinline__ void async_cp_b128(uint32_t lds_byte, uint32_t g_byte,
                                              const void* sbase) {
    asm volatile("global_load_async_to_lds_b128 %0, %1, %2"
                 :
                 : "v"(lds_byte), "v"(g_byte), "s"(sbase)
                 : "memory");
}

#define WAIT_ASYNCCNT(n) asm volatile("s_wait_asynccnt " #n ::: "memory")
#define WAIT_DSCNT0()    asm volatile("s_wait_dscnt 0x0"    ::: "memory")

__global__ __launch_bounds__(kBlock) void moe_reduce_rmsnorm_kernel(
    const float* __restrict__ residual,     // [T,H]
    const float* __restrict__ norm_weight,  // [H]
    const float* __restrict__ scale,        // [E,T]
    const float* __restrict__ active,       // [E,T,H]
    const float* __restrict__ token,        // [T,H]
    const float* __restrict__ eps_ptr,      // [1]
    float* __restrict__ out_hidden,         // [T,H]
    float* __restrict__ out_resid,          // [T,H]
    int T)
{
    __shared__ v4f   lds_buf[2][kRow4];     // 2 x 16 KB double buffer
    __shared__ float red[kBlock / 32];

    const int    t       = blockIdx.x;
    const int    tid     = threadIdx.x;
    const size_t row     = (size_t)t * kH;
    const size_t estride = (size_t)T * kH;  // expert stride in elements

    // Per-lane byte offsets (same for every expert row).
    uint32_t goff[kChunks];
    uint32_t loff[2][kChunks];
#pragma unroll
    for (int c = 0; c < kChunks; ++c) {
        const int idx = tid + c * kBlock;
        goff[c]    = (uint32_t)idx * 16u;
        loff[0][c] = (uint32_t)(uintptr_t)&lds_buf[0][idx];
        loff[1][c] = (uint32_t)(uintptr_t)&lds_buf[1][idx];
    }

    // Uniform per-block scalars (scalar loads).
    float s[kE];
#pragma unroll
    for (int e = 0; e < kE; ++e) s[e] = scale[(size_t)e * T + t];
    const float eps = eps_ptr[0];

    // Prologue: DMA expert 0 into buffer 0.
    {
        const float* base0 = active + row;
#pragma unroll
        for (int c = 0; c < kChunks; ++c) async_cp_b128(loff[0][c], goff[c], base0);
    }

    v4f acc[kChunks];
#pragma unroll
    for (int c = 0; c < kChunks; ++c) {
        v4f z = {0.0f, 0.0f, 0.0f, 0.0f};
        acc[c] = z;
    }

    // Double-buffered pipeline over experts.
#pragma unroll
    for (int e = 0; e < kE; ++e) {
        if (e + 1 < kE) {
            const float* nb = active + row + (size_t)(e + 1) * estride;
            WAIT_DSCNT0();  // buffer (e+1)&1 fully consumed before DMA refill
#pragma unroll
            for (int c = 0; c < kChunks; ++c)
                async_cp_b128(loff[(e + 1) & 1][c], goff[c], nb);
            WAIT_ASYNCCNT(0x4);  // async loads done in order -> expert e landed
        } else {
            WAIT_ASYNCCNT(0x0);
        }
        const float se = s[e];
        const int   b  = e & 1;
#pragma unroll
        for (int c = 0; c < kChunks; ++c) {
            v4f v = lds_buf[b][tid + c * kBlock];   // ds_load_b128
            acc[c] += se * v;
        }
    }

    // Shared-expert + residual add; write output_residual (stream-once -> NT).
    const v4f* tok4  = (const v4f*)(token    + row);
    const v4f* res4  = (const v4f*)(residual + row);
    v4f*       outr4 = (v4f*)(out_resid  + row);
    v4f*       outh4 = (v4f*)(out_hidden + row);
    const v4f* nw4   = (const v4f*)norm_weight;

    v4f   r[kChunks];
    float sumsq = 0.0f;
#pragma unroll
    for (int c = 0; c < kChunks; ++c) {
        const int idx = tid + c * kBlock;
        v4f a  = __builtin_nontemporal_load(&tok4[idx]);
        v4f b  = __builtin_nontemporal_load(&res4[idx]);
        v4f rv = acc[c] + a + b;
        r[c]   = rv;
        sumsq += rv.x * rv.x + rv.y * rv.y + rv.z * rv.z + rv.w * rv.w;
        __builtin_nontemporal_store(rv, &outr4[idx]);
    }

    // Block reduction: wave32 xor-shuffle tree, then 8-wave LDS combine.
#pragma unroll
    for (int off = 16; off > 0; off >>= 1)
        sumsq += __shfl_xor(sumsq, off, 32);
    if ((tid & 31) == 0) red[tid >> 5] = sumsq;
    __syncthreads();
    float tot = 0.0f;
#pragma unroll
    for (int w = 0; w < kBlock / 32; ++w) tot += red[w];

    const float inv = rsqrtf(tot * (1.0f / (float)kH) + eps);

    // RMSNorm output (norm_weight is tiny & reused -> regular loads, stays in L2).
#pragma unroll
    for (int c = 0; c < kChunks; ++c) {
        const int idx = tid + c * kBlock;
        v4f w = nw4[idx];
        v4f h = r[c] * inv * w;
        __builtin_nontemporal_store(h, &outh4[idx]);
    }
}

extern "C" void kernel_launch(void* const* d_in, const int* in_sizes, int n_in,
                              void* d_out, int out_size, void* d_ws, size_t ws_size,
                              hipStream_t stream) {
    // Input order: residual, norm_weight, device_num_experts, scale_input,
    //              active_experts_token_input, token_input, eps
    const float* residual    = (const float*)d_in[0];
    const float* norm_weight = (const float*)d_in[1];
    const float* scale       = (const float*)d_in[3];
    const float* active      = (const float*)d_in[4];
    const float* token       = (const float*)d_in[5];
    const float* eps         = (const float*)d_in[6];

    const int T = in_sizes[0] / kH;  // residual is [T,H]

    float* out_hidden = (float*)d_out;                        // output 0: [T,H]
    float* out_resid  = (float*)d_out + (size_t)T * kH;       // output 1: [T,H]

    hipLaunchKernelGGL(moe_reduce_rmsnorm_kernel,
                       dim3(T), dim3(kBlock), 0, stream,
                       residual, norm_weight, scale, active, token, eps,
                       out_hidden, out_resid, T);
}